// CompositeRGCN_83958020702635
// MI455X (gfx1250) — compile-verified
//
#include <hip/hip_runtime.h>
#include <hip/hip_bf16.h>
#include <math.h>

typedef __attribute__((ext_vector_type(2))) float v2f;
typedef __attribute__((ext_vector_type(8))) float v8f;

#define BB 512
#define NN 32
#define DD 300
#define RR 6
#define EE 256
#define NG 50000
#define NS 25000
#define KC 2112   // 7*300 = 2100 padded to mult of 16
#define WN 304    // padded inner N (300 -> 304 = 19*16)
#define NT2 19    // WN/16

// ---------------------------------------------------------------------------
// Kernel 1: per-sample edge coefficients + node-row mixing.
// Ycat[b, r*300 + d] = sum_n c[b,r,n] * x[b,n,d]   (r = 0..5)
// Ycat[b, 6*300 + d] = x[b,0,d]                    (for W0 self term)
// ---------------------------------------------------------------------------
__global__ void k_coeff_combine(const float* __restrict__ x,
                                const int*   __restrict__ edge_index,
                                const int*   __restrict__ edge_type,
                                float*       __restrict__ Ycat)
{
    int b   = blockIdx.x;
    int tid = threadIdx.x;                 // 320 threads = 10 waves
    __shared__ float deg[RR * NN];
    __shared__ float c[RR * NN];
    if (tid < RR * NN) { deg[tid] = 1.0f; c[tid] = 0.0f; }
    __syncthreads();

    int et = 0, src = 0, dst = 0;
    if (tid < EE) {
        et  = edge_type [b * EE + tid];
        src = edge_index[b * 2 * EE + tid];          // ei[0] = src
        dst = edge_index[b * 2 * EE + EE + tid];     // ei[1] = dst
        atomicAdd(&deg[et * NN + dst], 1.0f);        // deg = 1 + in-degree
    }
    __syncthreads();

    if (tid < RR) atomicAdd(&c[tid * NN], 1.0f / deg[tid * NN]);   // self-loop term
    if (tid < EE && dst == 0) {
        atomicAdd(&c[et * NN + src], rsqrtf(deg[et * NN + src] * deg[et * NN]));
    }
    __syncthreads();

    if (tid < DD) {
        const float* xb = x + (size_t)b * NN * DD;
        float acc[RR];
        #pragma unroll
        for (int r = 0; r < RR; ++r) acc[r] = 0.0f;
        for (int n = 0; n < NN; ++n) {
            float xv = xb[n * DD + tid];             // coalesced across tid
            #pragma unroll
            for (int r = 0; r < RR; ++r) acc[r] += c[r * NN + n] * xv;
        }
        float* yb = Ycat + (size_t)b * KC;
        #pragma unroll
        for (int r = 0; r < RR; ++r) yb[r * DD + tid] = acc[r];
        yb[RR * DD + tid] = xb[tid];                 // x[b,0,:] row
    }
    if (tid < KC - 7 * DD) Ycat[(size_t)b * KC + 7 * DD + tid] = 0.0f;  // K pad
}

// ---------------------------------------------------------------------------
// Kernel 1b: build Wcat [KC=2112, WN=304] = rowwise concat(W_rel[0..5], W0),
// zero padded in both K (2100->2112) and N (300->304).
// ---------------------------------------------------------------------------
__global__ void k_build_wcat(const float* __restrict__ W_rel,
                             const float* __restrict__ W0,
                             float*       __restrict__ Wcat)
{
    int idx = blockIdx.x * blockDim.x + threadIdx.x;
    if (idx >= KC * WN) return;
    int k = idx / WN, j = idx % WN;
    float v = 0.0f;
    if (j < DD && k < 7 * DD) {
        int r = k / DD, kr = k % DD;
        v = (r < RR) ? W_rel[((size_t)r * DD + kr) * DD + j]
                     : W0[(size_t)kr * DD + j];
    }
    Wcat[idx] = v;
}

// ---------------------------------------------------------------------------
// Kernel 2: H0 = Ycat[512,2112] @ Wcat[2112,304], leaky_relu(0.1) -> X0act.
// One 16x16 tile per wave via V_WMMA_F32_16X16X4_F32.
// ---------------------------------------------------------------------------
__global__ void k_gemm_h0(const float* __restrict__ Ycat,
                          const float* __restrict__ Wcat,
                          float*       __restrict__ X0act)
{
    int wave = threadIdx.x >> 5;
    int lane = threadIdx.x & 31;
    int tile = blockIdx.x * (blockDim.x >> 5) + wave;
    if (tile >= 32 * NT2) return;                 // wave-uniform guard
    int mt = tile / NT2, nt = tile % NT2;
    int l16 = lane & 15, half = lane >> 4;

    const float* arow = Ycat + (size_t)(mt * 16 + l16) * KC;
    const float* bcol = Wcat + nt * 16 + l16;

    v8f acc = {};
    for (int k0 = 0; k0 < KC; k0 += 4) {
        int ka = k0 + half * 2;                   // lanes 0-15: K{0,1}; 16-31: K{2,3}
        v2f a, bf;
        a.x  = arow[ka];
        a.y  = arow[ka + 1];
        bf.x = bcol[(size_t)ka * WN];
        bf.y = bcol[(size_t)(ka + 1) * WN];
        acc = __builtin_amdgcn_wmma_f32_16x16x4_f32(
                  false, a, false, bf, (short)0, acc, false, false);
    }
    #pragma unroll
    for (int j = 0; j < 8; ++j) {
        int m = mt * 16 + half * 8 + j;
        int n = nt * 16 + l16;
        float v = acc[j];
        v = (v >= 0.0f) ? v : 0.1f * v;           // leaky_relu(0.1)
        X0act[(size_t)m * WN + n] = v;
    }
}

// ---------------------------------------------------------------------------
// Kernel 3: logits = X0act[512, K=300] @ W[300, Ncols] + bias.
// Register-blocked WMMA: each wave computes 2 M-tiles x 4 N-tiles (32x64),
// 8 f32 WMMAs per K-step; A frags reused 4x, B frags reused 2x.
// ---------------------------------------------------------------------------
__global__ void k_gemm_logits(const float* __restrict__ X0act,
                              const float* __restrict__ W,
                              const float* __restrict__ bias,
                              float*       __restrict__ out,
                              int Ncols, int Ngroups)
{
    int wave = threadIdx.x >> 5;
    int lane = threadIdx.x & 31;
    int ng = blockIdx.x * (blockDim.x >> 5) + wave;   // N group: 64 columns
    if (ng >= Ngroups) return;                        // wave-uniform guard
    int my = blockIdx.y;                              // M group: 32 rows
    int l16 = lane & 15, half = lane >> 4;

    const float* arow0 = X0act + (size_t)(my * 32 + l16) * WN;
    const float* arow1 = arow0 + (size_t)16 * WN;
    int col0 = ng * 64 + l16;

    v8f acc[2][4] = {};
    for (int k0 = 0; k0 < DD; k0 += 4) {              // 300 % 4 == 0
        int ka = k0 + half * 2;                       // lanes 0-15: K{0,1}; 16-31: K{2,3}
        v2f a[2], bf[4];
        a[0].x = arow0[ka];
        a[0].y = arow0[ka + 1];
        a[1].x = arow1[ka];
        a[1].y = arow1[ka + 1];
        const float* bk = W + (size_t)ka * Ncols;
        __builtin_prefetch(bk + (size_t)8 * Ncols + col0, 0, 1);  // B stream ahead
        #pragma unroll
        for (int t = 0; t < 4; ++t) {
            int ct = col0 + t * 16;
            bool cv = (ct < Ncols);                   // per-lane -> selects only
            bf[t].x = cv ? bk[ct] : 0.0f;
            bf[t].y = cv ? bk[Ncols + ct] : 0.0f;
        }
        #pragma unroll
        for (int mi = 0; mi < 2; ++mi)
            #pragma unroll
            for (int t = 0; t < 4; ++t)
                acc[mi][t] = __builtin_amdgcn_wmma_f32_16x16x4_f32(
                    false, a[mi], false, bf[t], (short)0, acc[mi][t], false, false);
    }

    #pragma unroll
    for (int t = 0; t < 4; ++t) {
        int col = col0 + t * 16;
        if (col < Ncols) {
            float bv = bias[col];
            #pragma unroll
            for (int mi = 0; mi < 2; ++mi) {
                int mbase = my * 32 + mi * 16 + half * 8;
                #pragma unroll
                for (int j = 0; j < 8; ++j)
                    out[(size_t)(mbase + j) * Ncols + col] = acc[mi][t][j] + bv;
            }
        }
    }
}

// ---------------------------------------------------------------------------
// Kernel 4: in-place row-wise log_softmax. One block per row.
// ---------------------------------------------------------------------------
__global__ void k_log_softmax(float* __restrict__ logits, int Ncols)
{
    int b = blockIdx.x;
    float* row = logits + (size_t)b * Ncols;
    __shared__ float red[256];
    int tid = threadIdx.x;

    float mx = -3.402823466e38f;
    for (int j = tid; j < Ncols; j += blockDim.x) mx = fmaxf(mx, row[j]);
    red[tid] = mx; __syncthreads();
    for (int s = blockDim.x >> 1; s > 0; s >>= 1) {
        if (tid < s) red[tid] = fmaxf(red[tid], red[tid + s]);
        __syncthreads();
    }
    mx = red[0]; __syncthreads();

    float sum = 0.0f;
    for (int j = tid; j < Ncols; j += blockDim.x) sum += __expf(row[j] - mx);
    red[tid] = sum; __syncthreads();
    for (int s = blockDim.x >> 1; s > 0; s >>= 1) {
        if (tid < s) red[tid] += red[tid + s];
        __syncthreads();
    }
    float lse = mx + __logf(red[0]);
    __syncthreads();

    for (int j = tid; j < Ncols; j += blockDim.x) row[j] -= lse;
}

// ---------------------------------------------------------------------------
extern "C" void kernel_launch(void* const* d_in, const int* in_sizes, int n_in,
                              void* d_out, int out_size, void* d_ws, size_t ws_size,
                              hipStream_t stream)
{
    const float* x          = (const float*)d_in[0];
    const int*   edge_index = (const int*)  d_in[1];
    const int*   edge_type  = (const int*)  d_in[2];
    const float* W_rel      = (const float*)d_in[3];
    const float* W0         = (const float*)d_in[4];
    const float* Wg         = (const float*)d_in[5];
    const float* bg         = (const float*)d_in[6];
    const float* Ws         = (const float*)d_in[7];
    const float* bs         = (const float*)d_in[8];

    float* out_g = (float*)d_out;                       // [512, 50000]
    float* out_s = out_g + (size_t)BB * NG;             // [512, 25000]

    float* Ycat  = (float*)d_ws;                        // [512, 2112]
    float* Wcat  = Ycat + (size_t)BB * KC;              // [2112, 304]
    float* X0act = Wcat + (size_t)KC * WN;              // [512, 304]

    k_coeff_combine<<<BB, 320, 0, stream>>>(x, edge_index, edge_type, Ycat);

    int totW = KC * WN;
    k_build_wcat<<<(totW + 255) / 256, 256, 0, stream>>>(W_rel, W0, Wcat);

    k_gemm_h0<<<(32 * NT2 + 7) / 8, 256, 0, stream>>>(Ycat, Wcat, X0act);

    // logits GEMMs: grid.y = 16 M-groups (32 rows each), grid.x covers N-groups
    int ntg  = (NG + 15) / 16;                          // 3125 N tiles
    int ngg  = (ntg + 3) / 4;                           // 782 N groups (64 cols)
    dim3 g3g((ngg + 7) / 8, 16);
    k_gemm_logits<<<g3g, 256, 0, stream>>>(X0act, Wg, bg, out_g, NG, ngg);

    int nts  = (NS + 15) / 16;                          // 1563 N tiles
    int ngs  = (nts + 3) / 4;                           // 391 N groups
    dim3 g3s((ngs + 7) / 8, 16);
    k_gemm_logits<<<g3s, 256, 0, stream>>>(X0act, Ws, bs, out_s, NS, ngs);

    k_log_softmax<<<BB, 256, 0, stream>>>(out_g, NG);
    k_log_softmax<<<BB, 256, 0, stream>>>(out_s, NS);
}